// QuantumConvolution_33148557591179
// MI455X (gfx1250) — compile-verified
//
#include <hip/hip_runtime.h>

// Problem constants (from the reference)
#define BATCH 32
#define CIN   8
#define COUT  16
#define HW    32
#define ITDIM 30
#define NPIX  900     // 30*30
#define NLAY  3       // DRC
#define NQ    3

#define NMAT   (COUT * CIN * NLAY)       // 384 layer matrices
#define MATFL  (NMAT * 256)              // floats used for matrices in ws
// scaled params live right after the matrices in d_ws (1152 floats)

typedef float v2f __attribute__((ext_vector_type(2)));
typedef float v8f __attribute__((ext_vector_type(8)));

// ---------------------------------------------------------------------------
// Kernel 1: precompute, for each (cout, cin, layer):
//  (a) the constant circuit part M = P_cnot_ring * (Rot_q0 (x) Rot_q1 (x) Rot_q2)
//      as a REAL 16x16 matrix on [Re a0..a7 ; Im a0..a7], row-major (256 f32)
//  (b) pre-scaled RY params: param * 0.5 (half angle) * 1/(2*pi) (v_sin_f32
//      operates in "turns"), so the main kernel needs 1 mul + raw v_sin/v_cos.
// ws usage: 384*256*4 + 1152*4 = 397,824 bytes.
// ---------------------------------------------------------------------------
__global__ void build_layer_mats(const float* __restrict__ weights,
                                 const float* __restrict__ params,
                                 float* __restrict__ ws) {
    int t = blockIdx.x * blockDim.x + threadIdx.x;
    if (t >= NMAT) return;
    int l    = t % NLAY;
    int cin  = (t / NLAY) % CIN;
    int cout = t / (NLAY * CIN);

    // (b) scaled RY params: layout matches t ordering exactly
    {
        const float kscale = 0.25f / 3.14159265358979323846f;  // 0.5 / (2*pi)
        float* ps = ws + MATFL;
        #pragma unroll
        for (int q = 0; q < NQ; ++q)
            ps[t * NQ + q] = params[t * NQ + q] * kscale;
    }

    // (a) Rot(phi, theta, omega) complex 2x2 per qubit
    float Ur[3][2][2], Ui[3][2][2];
    for (int q = 0; q < 3; ++q) {
        const float* w = weights + ((((cout * CIN + cin) * NLAY + l) * NQ + q) * 3);
        float phi = w[0], th = w[1], om = w[2];
        float c  = cosf(0.5f * th), s = sinf(0.5f * th);
        float apo = 0.5f * (phi + om), amo = 0.5f * (phi - om);
        float epr = cosf(apo), epi = -sinf(apo);   // exp(-i(phi+om)/2)
        float emr = cosf(amo), emi =  sinf(amo);   // exp(+i(phi-om)/2)
        Ur[q][0][0] =  epr * c;  Ui[q][0][0] =  epi * c;   // ep*c
        Ur[q][0][1] = -emr * s;  Ui[q][0][1] = -emi * s;   // -em*s
        Ur[q][1][0] =  emr * s;  Ui[q][1][0] = -emi * s;   // conj(em)*s
        Ur[q][1][1] =  epr * c;  Ui[q][1][1] = -epi * c;   // conj(ep)*c
    }

    float* M = ws + t * 256;
    for (int i = 0; i < 8; ++i) {
        // CNOT ring folded in: out_row i reads product-matrix row g(i),
        // g(a,b,c) = (a^c, a^b^c, b^c)   [a = bit2 = qubit0]
        int a = (i >> 2) & 1, b = (i >> 1) & 1, cb = i & 1;
        int gi = ((a ^ cb) << 2) | ((a ^ b ^ cb) << 1) | (b ^ cb);
        int ga = (gi >> 2) & 1, gb = (gi >> 1) & 1, gc = gi & 1;
        for (int j = 0; j < 8; ++j) {
            int ja = (j >> 2) & 1, jb = (j >> 1) & 1, jc = j & 1;
            float r0 = Ur[0][ga][ja], i0 = Ui[0][ga][ja];
            float r1 = Ur[1][gb][jb], i1 = Ui[1][gb][jb];
            float r2 = Ur[2][gc][jc], i2 = Ui[2][gc][jc];
            float rr = r0 * r1 - i0 * i1;
            float ri = r0 * i1 + i0 * r1;
            float cr = rr * r2 - ri * i2;
            float ci = rr * i2 + ri * r2;
            // real 16x16: [[Re, -Im], [Im, Re]]
            M[i * 16 + j]           =  cr;
            M[i * 16 + j + 8]       = -ci;
            M[(i + 8) * 16 + j]     =  ci;
            M[(i + 8) * 16 + j + 8] =  cr;
        }
    }
}

// ---------------------------------------------------------------------------
// Lane-half swap (lane i <-> lane i^16) on the VALU: v_permlanex16_b32 with
// identity nibble selects. Avoids ds_bpermute + s_wait_dscnt in the hot loop.
// ---------------------------------------------------------------------------
__device__ __forceinline__ float swap16(float x) {
    int i = __float_as_int(x);
    int r = __builtin_amdgcn_permlanex16(i, i, 0x76543210, 0xFEDCBA98,
                                         /*fi=*/false, /*bound_ctrl=*/false);
    return __int_as_float(r);
}

__device__ __forceinline__ v8f wmma4(v2f a, v2f b, v8f c) {
    return __builtin_amdgcn_wmma_f32_16x16x4_f32(
        /*neg_a=*/false, a, /*neg_b=*/false, b,
        /*c_mod=*/(short)0, c, /*reuse_a=*/false, /*reuse_b=*/false);
}

// ---------------------------------------------------------------------------
// Main kernel.  One wave simulates 16 pixel-states in WMMA D-layout:
//   lane L (0..15):  components 0..7  (Re a0..a7) of state L in S[0..7]
//   lane L+16:       components 8..15 (Im a0..a7) of state L in S[0..7]
// RY gates: per-lane real 2x2 rotations across VGPR pairs (no shuffles),
// angles via pre-scaled params + raw v_sin_f32/v_cos_f32 (turns).
// Fixed layer matrix: chained v_wmma_f32_16x16x4_f32 (A from LDS,
// B rebuilt from the state with permlanex16 half-swaps).
// Layer 0 specialization: |000> -> separable REAL product state, so only the
// two WMMA chunks touching real components are needed.
// ---------------------------------------------------------------------------
__global__ __launch_bounds__(256) void qconv_main(
    const float* __restrict__ inputs,    // (32, 8, 32, 32)
    const float* __restrict__ pscaled,   // (16, 8, 3, 3) pre-scaled
    const float* __restrict__ mats,      // (16, 8, 3) x 16x16
    float* __restrict__ out)             // (32, 16, 30, 30)
{
    __shared__ float smat[CIN * NLAY * 256];   // 24 KB: all matrices of one cout

    const int cout = blockIdx.y;
    const int bidx = blockIdx.z;

    // Cooperative stage of this cout's 24 layer matrices into LDS
    const float* src = mats + cout * (CIN * NLAY * 256);
    for (int i = threadIdx.x; i < CIN * NLAY * 256; i += blockDim.x)
        smat[i] = src[i];
    __syncthreads();

    const int wave = threadIdx.x >> 5;
    const int lane = threadIdx.x & 31;
    const int half = lane >> 4;       // 0 = Re half, 1 = Im half
    const int ln   = lane & 15;       // state index within wave tile
    const int tile = blockIdx.x * 8 + wave;
    if (tile * 16 >= NPIX) return;    // wave-uniform

    int  p     = tile * 16 + ln;
    bool valid = p < NPIX;
    int  pc    = valid ? p : (NPIX - 1);   // clamp: keep EXEC all-ones for WMMA
    int  oi    = pc / ITDIM, oj = pc % ITDIM;

    float acc = 0.0f;
    for (int cin = 0; cin < CIN; ++cin) {
        const float* ip = inputs + ((size_t)(bidx * CIN + cin) * HW + oi) * HW + oj;
        if (cin + 1 < CIN)  // hint next input plane into cache (global_prefetch_b8)
            __builtin_prefetch(ip + HW * HW, 0, 0);

        // 3x3 window pixels (layer l = window row, qubit q = window col)
        float x[3][3];
        #pragma unroll
        for (int l = 0; l < 3; ++l)
            #pragma unroll
            for (int q = 0; q < 3; ++q)
                x[l][q] = ip[l * HW + q];

        const float* pp  = pscaled + (size_t)(cout * CIN + cin) * (NLAY * NQ);
        const float* mb0 = smat + cin * (NLAY * 256);

        v8f S;
        v2f a, b;

        // ---------------- layer 0 (specialized) ----------------
        {
            float t0 = x[0][0] * pp[0];        // angle in "turns", half folded
            float t1 = x[0][1] * pp[1];
            float t2 = x[0][2] * pp[2];
            float c0 = __builtin_amdgcn_cosf(t0), s0 = __builtin_amdgcn_sinf(t0);
            float c1 = __builtin_amdgcn_cosf(t1), s1 = __builtin_amdgcn_sinf(t1);
            float c2 = __builtin_amdgcn_cosf(t2), s2 = __builtin_amdgcn_sinf(t2);
            // zero the imaginary half directly
            float c2m = half ? 0.0f : c2;
            float s2m = half ? 0.0f : s2;
            float cc = c0 * c1, cs = c0 * s1, sc = s0 * c1, ss = s0 * s1;
            S[0] = cc * c2m;  S[1] = cc * s2m;
            S[2] = cs * c2m;  S[3] = cs * s2m;
            S[4] = sc * c2m;  S[5] = sc * s2m;
            S[6] = ss * c2m;  S[7] = ss * s2m;

            // Imag components are exactly 0 => WMMA chunks 2,3 contribute 0.
            float sh2 = swap16(S[2]), sh3 = swap16(S[3]);
            float sh6 = swap16(S[6]), sh7 = swap16(S[7]);

            const float* m = mb0 + (ln * 16 + half * 2);  // layer 0 matrix
            v8f d;
            #pragma unroll
            for (int i = 0; i < 8; ++i) d[i] = 0.0f;

            a = *(const v2f*)(m + 0);
            b[0] = half ? sh2 : S[0];  b[1] = half ? sh3 : S[1];
            d = wmma4(a, b, d);
            a = *(const v2f*)(m + 4);
            b[0] = half ? sh6 : S[4];  b[1] = half ? sh7 : S[5];
            d = wmma4(a, b, d);
            S = d;
        }

        // ---------------- layers 1, 2 (generic) ----------------
        #pragma unroll
        for (int l = 1; l < 3; ++l) {
            #pragma unroll
            for (int q = 0; q < 3; ++q) {
                float th = x[l][q] * pp[l * 3 + q];   // turns
                float c = __builtin_amdgcn_cosf(th);
                float s = __builtin_amdgcn_sinf(th);
                const int stride = 4 >> q;          // qubit0:4, qubit1:2, qubit2:1
                #pragma unroll
                for (int base = 0; base < 8; ++base) {
                    if (base & stride) continue;
                    float a0 = S[base], a1 = S[base + stride];
                    S[base]          = c * a0 - s * a1;
                    S[base + stride] = s * a0 + c * a1;
                }
            }

            // cross-half copies for B-operand construction (VALU permlanex16)
            float sh0 = swap16(S[0]), sh1 = swap16(S[1]);
            float sh2 = swap16(S[2]), sh3 = swap16(S[3]);
            float sh4 = swap16(S[4]), sh5 = swap16(S[5]);
            float sh6 = swap16(S[6]), sh7 = swap16(S[7]);

            // A layout (16x4 f32): vgpr0 = col 4c (+2 in upper half), vgpr1 = +1
            const float* m = mb0 + l * 256 + (ln * 16 + half * 2); // 8B aligned
            v8f d;
            #pragma unroll
            for (int i = 0; i < 8; ++i) d[i] = 0.0f;

            // chunk 0: state rows k = 0..3
            a = *(const v2f*)(m + 0);
            b[0] = half ? sh2 : S[0];  b[1] = half ? sh3 : S[1];
            d = wmma4(a, b, d);
            // chunk 1: k = 4..7
            a = *(const v2f*)(m + 4);
            b[0] = half ? sh6 : S[4];  b[1] = half ? sh7 : S[5];
            d = wmma4(a, b, d);
            // chunk 2: k = 8..11
            a = *(const v2f*)(m + 8);
            b[0] = half ? S[2] : sh0;  b[1] = half ? S[3] : sh1;
            d = wmma4(a, b, d);
            // chunk 3: k = 12..15
            a = *(const v2f*)(m + 12);
            b[0] = half ? S[6] : sh4;  b[1] = half ? S[7] : sh5;
            d = wmma4(a, b, d);

            S = d;
        }

        // <Z2>: sum_i sign(i) * amp_i^2, sign = +1 if last qubit bit == 0.
        // Each half contributes its (Re^2 or Im^2) part; combine across halves.
        float w = S[0] * S[0] - S[1] * S[1] + S[2] * S[2] - S[3] * S[3]
                + S[4] * S[4] - S[5] * S[5] + S[6] * S[6] - S[7] * S[7];
        w += swap16(w);
        acc += w;
    }

    if (!half && valid)
        out[(size_t)(bidx * COUT + cout) * NPIX + p] = acc;
}

// ---------------------------------------------------------------------------
extern "C" void kernel_launch(void* const* d_in, const int* in_sizes, int n_in,
                              void* d_out, int out_size, void* d_ws, size_t ws_size,
                              hipStream_t stream) {
    const float* inputs  = (const float*)d_in[0];  // (32,8,32,32)
    const float* params  = (const float*)d_in[1];  // (16,8,3,3)
    const float* weights = (const float*)d_in[2];  // (16,8,3,3,3)
    float*       out     = (float*)d_out;          // (32,16,30,30)
    float*       ws      = (float*)d_ws;           // 384KB mats + 4.5KB params

    // 1) build the 384 constant 16x16 layer matrices + scaled RY params
    hipLaunchKernelGGL(build_layer_mats, dim3(3), dim3(128), 0, stream,
                       weights, params, ws);

    // 2) main quantum-conv kernel: grid = (pixel-tiles/8, cout, batch)
    hipLaunchKernelGGL(qconv_main, dim3(8, COUT, BATCH), dim3(256), 0, stream,
                       inputs, ws + MATFL, ws, out);
}